// Hawkes_31963146616942
// MI455X (gfx1250) — compile-verified
//
#include <hip/hip_runtime.h>

// Hawkes intensity evaluation, restructured around the 16-mark algebra:
//   G[m,k]   = sum_{i: marks[i]==m, mask[i]} exp(-Alpha[m,k]*dist[i])
//   out[s,k] = mu[k] + sum_m (A[m,k]*G[m,k]) * exp(-Alpha[m,k]*dts[s])
// Phase 2 does the sum over m with V_WMMA_F32_16X16X4_F32 (Ones(16x4) x H(4x16)).

typedef __attribute__((ext_vector_type(2))) float v2f;
typedef __attribute__((ext_vector_type(8))) float v8f;

#define T_EVENTS   16384
#define S_QUERIES  512
#define K_MARKS    16
#define NBLK_G     32
#define CHUNK_G    (T_EVENTS / NBLK_G)   // 512 events per block in phase 1

// ---------------------------------------------------------------------------
// Kernel 1: dist[i] = suffix sum of masked gaps. One block, 256 threads,
// 64 consecutive indices per thread. All summation orders are fixed ->
// deterministic across replays.
// ---------------------------------------------------------------------------
__global__ void __launch_bounds__(256)
hawkes_dist_kernel(const float* __restrict__ ts,
                   const unsigned char* __restrict__ mask,
                   float* __restrict__ dist) {
    __shared__ float csum[256];
    const int t  = threadIdx.x;
    const int lo = t * 64;

    float s = 0.0f;
    for (int i = lo; i < lo + 64; ++i) {
        float g = 0.0f;
        if (i < T_EVENTS - 1 && mask[i + 1]) g = ts[i + 1] - ts[i];
        s += g;
    }
    csum[t] = s;
    __syncthreads();

    float suffix = 0.0f;
    for (int u = t + 1; u < 256; ++u) suffix += csum[u];

    float running = suffix;
    for (int i = lo + 63; i >= lo; --i) {
        float g = 0.0f;
        if (i < T_EVENTS - 1 && mask[i + 1]) g = ts[i + 1] - ts[i];
        running += g;
        dist[i] = running;   // inclusive suffix sum; dist[T-1] == 0
    }
}

// ---------------------------------------------------------------------------
// Kernel 2: per-block partial G. Thread t owns cell (m,k) = (t>>4, t&15) and
// scans a contiguous event range in order (deterministic). 32 blocks.
// ---------------------------------------------------------------------------
__global__ void __launch_bounds__(256)
hawkes_gpartial_kernel(const float* __restrict__ dist,
                       const int* __restrict__ marks,
                       const unsigned char* __restrict__ mask,
                       const float* __restrict__ Alpha,
                       float* __restrict__ gpart) {
    const int t = threadIdx.x;          // 0..255 -> (m,k)
    const int m = t >> 4;
    const float nalpha = -Alpha[t];     // Alpha[m*16 + k]

    const int lo = blockIdx.x * CHUNK_G;
    const int hi = lo + CHUNK_G;

    float acc = 0.0f;
    for (int i = lo; i < hi; ++i) {
        const int   mi = marks[i];
        const float d  = dist[i];
        const float e  = __expf(nalpha * d);     // v_exp_f32
        const bool  on = (mi == m) && (mask[i] != 0);
        acc += on ? e : 0.0f;
    }
    gpart[blockIdx.x * 256 + t] = acc;
}

// ---------------------------------------------------------------------------
// Kernel 3: fold block partials in fixed order, fuse P[m,k] = A[m,k]*G[m,k].
// ---------------------------------------------------------------------------
__global__ void __launch_bounds__(256)
hawkes_reduce_kernel(const float* __restrict__ gpart,
                     const float* __restrict__ A,
                     float* __restrict__ P) {
    const int t = threadIdx.x;
    float acc = 0.0f;
    for (int b = 0; b < NBLK_G; ++b) acc += gpart[b * 256 + t];
    P[t] = A[t] * acc;
}

// ---------------------------------------------------------------------------
// Kernel 4: one wave per query s. Lane L holds column k = L&15 of the 16x16
// matrix H[m,k] = P[m,k]*exp(-Alpha[m,k]*dt). Column sums over m computed as
// D += Ones(16x4) x H_chunk(4x16) with V_WMMA_F32_16X16X4_F32, 4 chunks.
// B-matrix f32 16x16x4 layout: VGPR0 = {rows K=0 | K=2}, VGPR1 = {K=1 | K=3}
// across lane halves -> slot r of chunk c is m = 4c + 2*(lane>=16) + r.
// Result row M=0 lives in accumulator element 0, lanes 0..15.
// ---------------------------------------------------------------------------
__global__ void __launch_bounds__(256)
hawkes_intensity_kernel(const float* __restrict__ dts,
                        const float* __restrict__ Alpha,
                        const float* __restrict__ mu,
                        const float* __restrict__ P,
                        float* __restrict__ out) {
    const int lane = threadIdx.x & 31;
    const int wave = (blockIdx.x * blockDim.x + threadIdx.x) >> 5;  // = s
    const float dt = dts[wave];

    const int k    = lane & 15;
    const int half = lane >> 4;

    const v2f ones = {1.0f, 1.0f};
    v8f d = {};

#pragma unroll
    for (int c = 0; c < 4; ++c) {
        const int m0 = 4 * c + 2 * half + 0;
        const int m1 = 4 * c + 2 * half + 1;
        v2f b;
        b.x = P[m0 * 16 + k] * __expf(-Alpha[m0 * 16 + k] * dt);
        b.y = P[m1 * 16 + k] * __expf(-Alpha[m1 * 16 + k] * dt);
        // 8 args: (neg_a, A, neg_b, B, c_mod, C, reuse_a, reuse_b)
        d = __builtin_amdgcn_wmma_f32_16x16x4_f32(false, ones, false, b,
                                                  (short)0, d, false, false);
    }

    if (lane < 16) out[wave * K_MARKS + k] = mu[k] + d[0];
}

// ---------------------------------------------------------------------------
extern "C" void kernel_launch(void* const* d_in, const int* in_sizes, int n_in,
                              void* d_out, int out_size, void* d_ws, size_t ws_size,
                              hipStream_t stream) {
    const float*         ts    = (const float*)d_in[0];
    const int*           marks = (const int*)d_in[1];
    const unsigned char* mask  = (const unsigned char*)d_in[2];  // jax bool = 1 byte
    const float*         dts   = (const float*)d_in[3];
    const float*         A     = (const float*)d_in[4];
    const float*         Alpha = (const float*)d_in[5];
    const float*         mu    = (const float*)d_in[6];
    float*               out   = (float*)d_out;

    // Workspace layout (floats): dist[T] | gpart[NBLK_G*256] | P[256]
    float* dist  = (float*)d_ws;
    float* gpart = dist + T_EVENTS;
    float* P     = gpart + NBLK_G * 256;

    hawkes_dist_kernel<<<1, 256, 0, stream>>>(ts, mask, dist);
    hawkes_gpartial_kernel<<<NBLK_G, 256, 0, stream>>>(dist, marks, mask, Alpha, gpart);
    hawkes_reduce_kernel<<<1, 256, 0, stream>>>(gpart, A, P);
    hawkes_intensity_kernel<<<S_QUERIES / 8, 256, 0, stream>>>(dts, Alpha, mu, P, out);
}